// AttnApply_21328807592051
// MI455X (gfx1250) — compile-verified
//
#include <hip/hip_runtime.h>

// AttnApply as banded matmul on CDNA5 WMMA (fp32, V_WMMA_F32_16X16X4_F32).
// out[b,t,c] = sum_i X[b, t+i-5, c] * W[b,t,i],  T=11, D=5.
//
// Each wave processes TWO adjacent 16-row output tiles (t0, t0+16) for its
// c-tiles: their K-windows overlap by 16 rows, so the 12 loaded row-pair
// registers R[0..11] serve tile1 (R[0..7]) and tile2 (R[4..11]) -> X read
// amplification drops 2.0x -> 1.5x, and the two accumulator chains are
// independent, overlapping WMMA latency.
// Interior blocks are branchless; boundary blocks (first/last) use
// clamp+select (v_cndmask), never exec-mask branches.
// Output is stored non-temporally to keep L2 for the reused X rows.

typedef __attribute__((ext_vector_type(2)))  float v2f;
typedef __attribute__((ext_vector_type(8)))  float v8f;

#define Bsz  32
#define Lsz  4096
#define Csz  256
#define Tw   11
#define Dpad 5

__device__ __forceinline__ int iclamp(int v, int lo, int hi) {
  return v < lo ? lo : (v > hi ? hi : v);
}

// A (16x32 band as 8 K-steps of 16x4): lane -> M = l16; step s holds
// K = 4s+2*half (x) and 4s+2*half+1 (y). Band mask is branchless.
__device__ __forceinline__ void load_A(const float* __restrict__ Wrow,
                                       int half, int l16, v2f (&A)[8]) {
#pragma unroll
  for (int s = 0; s < 8; ++s) {
    const int i0 = 4 * s + 2 * half - l16;
    const int i1 = i0 + 1;
    const float a0 = Wrow[iclamp(i0, 0, Tw - 1)];
    const float a1 = Wrow[iclamp(i1, 0, Tw - 1)];
    A[s].x = (i0 >= 0 && i0 < Tw) ? a0 : 0.0f;
    A[s].y = (i1 >= 0 && i1 < Tw) ? a1 : 0.0f;
  }
}

template <bool CLAMP>
__device__ __forceinline__ void compute_pair(
    const float* __restrict__ Wrow,  // W + (b*L + t0 + l16)*Tw
    const float* __restrict__ Xb,    // X + b*L*C
    float*       __restrict__ Ob,    // O + b*L*C
    int t0, int wave, int half, int l16)
{
  v2f A1[8], A2[8];
  load_A(Wrow,            half, l16, A1);   // rows t0   .. t0+15
  load_A(Wrow + 16 * Tw,  half, l16, A2);   // rows t0+16.. t0+31

  const int rbase = t0 - Dpad;  // K=0 input row for tile 1

#pragma unroll
  for (int ct = 0; ct < 4; ++ct) {
    const int c0 = wave * 64 + ct * 16;

    // 12 shared row-pair registers covering input rows [t0-5, t0+43).
    // B(4x16) step layout: lane -> N = l16; x = row 4s+2*half, y = row +1.
    v2f R[12];
#pragma unroll
    for (int s = 0; s < 12; ++s) {
      const int r0 = rbase + 4 * s + 2 * half;
      const int r1 = r0 + 1;
      if (CLAMP) {
        const float b0 = Xb[(unsigned)iclamp(r0, 0, Lsz - 1) * Csz + c0 + l16];
        const float b1 = Xb[(unsigned)iclamp(r1, 0, Lsz - 1) * Csz + c0 + l16];
        R[s].x = (r0 >= 0 && r0 < Lsz) ? b0 : 0.0f;
        R[s].y = (r1 >= 0 && r1 < Lsz) ? b1 : 0.0f;
      } else {
        R[s].x = Xb[(unsigned)r0 * Csz + c0 + l16];
        R[s].y = Xb[(unsigned)r1 * Csz + c0 + l16];
      }
    }

    // Two independent 8-deep wmma chains, interleaved.
    v8f acc1 = {}, acc2 = {};
#pragma unroll
    for (int s = 0; s < 8; ++s) {
      acc1 = __builtin_amdgcn_wmma_f32_16x16x4_f32(
          false, A1[s], false, R[s],     (short)0, acc1, false, false);
      acc2 = __builtin_amdgcn_wmma_f32_16x16x4_f32(
          false, A2[s], false, R[s + 4], (short)0, acc2, false, false);
    }

    // D(16x16): VGPR v -> row v + 8*half, col l16 (half-wave 64B segments).
    // Non-temporal: output is write-once, keep L2 for X.
#pragma unroll
    for (int v = 0; v < 8; ++v) {
      const int m = v + 8 * half;
      __builtin_nontemporal_store(
          acc1[v], &Ob[(unsigned)(t0 + m) * Csz + c0 + l16]);
      __builtin_nontemporal_store(
          acc2[v], &Ob[(unsigned)(t0 + 16 + m) * Csz + c0 + l16]);
    }
  }
}

__global__ __launch_bounds__(128) void attn_apply_wmma(
    const float* __restrict__ X,   // [B, L, C]
    const float* __restrict__ W,   // [B, L, T]
    float* __restrict__ O)         // [B, L, C]
{
  const int lane = threadIdx.x & 31;
  const int wave = threadIdx.x >> 5;
  const int half = lane >> 4;
  const int l16  = lane & 15;

  const int t0 = blockIdx.x * 32;    // pair of 16-row output tiles
  const int b  = blockIdx.y;

  const float* Wrow = W + ((size_t)b * Lsz + (size_t)(t0 + l16)) * Tw;
  const float* Xb   = X + (size_t)b * Lsz * Csz;
  float*       Ob   = O + (size_t)b * Lsz * Csz;

  // Interior iff input rows [t0-5, t0+43) all lie in [0, L): uniform/block.
  const bool interior = (t0 - Dpad >= 0) && (t0 + 48 - Dpad <= Lsz);
  if (interior) {
    compute_pair<false>(Wrow, Xb, Ob, t0, wave, half, l16);
  } else {
    compute_pair<true>(Wrow, Xb, Ob, t0, wave, half, l16);
  }
}

extern "C" void kernel_launch(void* const* d_in, const int* in_sizes, int n_in,
                              void* d_out, int out_size, void* d_ws, size_t ws_size,
                              hipStream_t stream) {
  const float* X = (const float*)d_in[0];  // [32, 4096, 256] fp32
  const float* W = (const float*)d_in[1];  // [32, 4096, 11]  fp32
  float* O = (float*)d_out;                // [32, 4096, 256] fp32

  dim3 grid(Lsz / 32, Bsz);                // 128 x 32 blocks
  attn_apply_wmma<<<grid, 128, 0, stream>>>(X, W, O);
}